// XDoGPass1Effect_47012712022490
// MI455X (gfx1250) — compile-verified
//
#include <hip/hip_runtime.h>
#include <cstdint>

// LIC-based XDoG pass 1 for MI455X (gfx1250).
// Strategy: tile 32x32 outputs per block; TDM (tensor_load_to_lds) stages a
// 64x64 window (tile + 16 halo, clamped inside the image) of tangent_x,
// tangent_y and x[ch0] into LDS; all streamline bilinear gathers then hit LDS.
// TENSORcnt + workgroup barrier for visibility. Whole working set (48MB) is
// L2-resident (192MB), so the TDM reads stream from L2 at near-L2 bandwidth.

#define W      1024
#define H      1024
#define NIMG   4
#define TILE   32
#define HALO   16
#define WIN    64              // TILE + 2*HALO
#define MAXSTEP 24             // dw >= ~0.5 per step, half_step <= 8 -> <=17 steps
#define DIV_EPS 1e-8f
#define FEPS    1.1920929e-7f  // float32 machine eps

typedef uint32_t u32x4 __attribute__((ext_vector_type(4)));
typedef int32_t  i32x4 __attribute__((ext_vector_type(4)));
typedef int32_t  i32x8 __attribute__((ext_vector_type(8)));

#if defined(__has_builtin)
#  if __has_builtin(__builtin_amdgcn_tensor_load_to_lds)
#    define HAVE_TDM 1
#  endif
#endif
#ifndef HAVE_TDM
#  define HAVE_TDM 0
#endif

__device__ __forceinline__ void tdm_load_2d(uint32_t lds_byte_off,
                                            const float* gptr,
                                            int tile_w, int tile_h,
                                            int stride_elems) {
#if HAVE_TDM
  uint64_t ga = (uint64_t)(uintptr_t)gptr;
  // ---- D# group 0 (128b): count=1 | lds_addr | global_addr | type=2 ----
  u32x4 g0;
  g0[0] = 1u;                                   // count=1, user mode, no gather
  g0[1] = lds_byte_off;                         // lds_addr
  g0[2] = (uint32_t)ga;                         // global_addr[31:0]
  g0[3] = (uint32_t)((ga >> 32) & 0x1FFFFFFu)   // global_addr[56:32]
        | (2u << 30);                           // type = 2 ("image")
  // ---- D# group 1 (256b) ----
  i32x8 g1;
  g1[0] = (int)(2u << 16);                      // wg_mask=0, data_size=2 (4B)
  g1[1] = (int)(((uint32_t)W & 0xFFFFu) << 16);               // tensor_dim0 lo16
  g1[2] = (int)((((uint32_t)W >> 16) & 0xFFFFu)
        | (((uint32_t)H & 0xFFFFu) << 16));                   // dim0 hi / dim1 lo
  g1[3] = (int)((((uint32_t)H >> 16) & 0xFFFFu)
        | (((uint32_t)tile_w & 0xFFFFu) << 16));              // dim1 hi / tile_dim0
  g1[4] = (int)((uint32_t)tile_h & 0xFFFFu);                  // tile_dim1, tile_dim2=0
  g1[5] = (int)(uint32_t)stride_elems;                        // tensor_dim0_stride[31:0]
  g1[6] = 0;                                                  // stride hi / dim1_stride lo
  g1[7] = 0;
  i32x4 g2 = {0, 0, 0, 0};
  i32x4 g3 = {0, 0, 0, 0};
#if defined(__clang_major__) && (__clang_major__ >= 23)
  i32x8 g4 = {0, 0, 0, 0, 0, 0, 0, 0};
  __builtin_amdgcn_tensor_load_to_lds(g0, g1, g2, g3, g4, 0);
#else
  __builtin_amdgcn_tensor_load_to_lds(g0, g1, g2, g3, 0);
#endif
#else
  (void)lds_byte_off; (void)gptr; (void)tile_w; (void)tile_h; (void)stride_elems;
#endif
}

__device__ __forceinline__ float sgnf(float v) {
  return (v > 0.0f) ? 1.0f : ((v < 0.0f) ? -1.0f : 0.0f);
}
__device__ __forceinline__ float safediv(float a, float b) {
  float bs = (fabsf(b) < DIV_EPS) ? ((b < 0.0f) ? -DIV_EPS : DIV_EPS) : b;
  return a / bs;
}

struct Bil { int i00, i01, i10, i11; float w00, w01, w10, w11; };

__device__ __forceinline__ Bil bil_idx(float px, float py, int w0x, int w0y) {
  float fx = px * (float)W - 0.5f;
  float fy = py * (float)H - 0.5f;
  float x0 = floorf(fx), y0 = floorf(fy);
  float wx = fx - x0,    wy = fy - y0;
  int xi = (int)x0, yi = (int)y0;
  int x0i = min(max(xi,     0), W - 1);
  int x1i = min(max(xi + 1, 0), W - 1);
  int y0i = min(max(yi,     0), H - 1);
  int y1i = min(max(yi + 1, 0), H - 1);
  // map clamped global texel -> window-local (window lies fully inside image;
  // extra clamp is belt-and-braces for degenerate huge-dw steps)
  int lx0 = min(max(x0i - w0x, 0), WIN - 1);
  int lx1 = min(max(x1i - w0x, 0), WIN - 1);
  int ly0 = min(max(y0i - w0y, 0), WIN - 1);
  int ly1 = min(max(y1i - w0y, 0), WIN - 1);
  Bil b;
  b.i00 = ly0 * WIN + lx0;  b.i01 = ly0 * WIN + lx1;
  b.i10 = ly1 * WIN + lx0;  b.i11 = ly1 * WIN + lx1;
  b.w00 = (1.0f - wx) * (1.0f - wy);
  b.w01 = wx * (1.0f - wy);
  b.w10 = (1.0f - wx) * wy;
  b.w11 = wx * wy;
  return b;
}

__global__ __launch_bounds__(256)
void lic_xdog_kernel(const float* __restrict__ x,
                     const float* __restrict__ tangent,
                     const float* __restrict__ epsilon,
                     const float* __restrict__ sigma_edge,
                     const float* __restrict__ phi,
                     float* __restrict__ out) {
  __shared__ float s_tx[WIN * WIN];
  __shared__ float s_ty[WIN * WIN];
  __shared__ float s_xc[WIN * WIN];

  const int n     = blockIdx.z;
  const int tileX = blockIdx.x * TILE;
  const int tileY = blockIdx.y * TILE;
  // staged window, clamped fully inside the image (W,H >= WIN)
  const int w0x = min(max(tileX - HALO, 0), W - WIN);
  const int w0y = min(max(tileY - HALO, 0), H - WIN);

  const size_t HW    = (size_t)H * W;
  const float* gtx = tangent + ((size_t)n * 2 + 0) * HW + (size_t)w0y * W + w0x;
  const float* gty = tangent + ((size_t)n * 2 + 1) * HW + (size_t)w0y * W + w0x;
  const float* gxc = x       + ((size_t)n * 3 + 0) * HW + (size_t)w0y * W + w0x;

#if HAVE_TDM
  if (threadIdx.x < 32) {   // wave 0 only: TDM issues regardless of EXEC
    tdm_load_2d((uint32_t)(uintptr_t)(void*)s_tx, gtx, WIN, WIN, W);
    tdm_load_2d((uint32_t)(uintptr_t)(void*)s_ty, gty, WIN, WIN, W);
    tdm_load_2d((uint32_t)(uintptr_t)(void*)s_xc, gxc, WIN, WIN, W);
#if defined(__has_builtin) && __has_builtin(__builtin_amdgcn_s_wait_tensorcnt)
    __builtin_amdgcn_s_wait_tensorcnt(0);
#else
    asm volatile("s_wait_tensorcnt 0x0" ::: "memory");
#endif
  }
#else
  for (int i = threadIdx.x; i < WIN * WIN; i += 256) {
    int ry = i >> 6, rx = i & (WIN - 1);
    size_t go = (size_t)ry * W + rx;
    s_tx[i] = gtx[go];
    s_ty[i] = gty[go];
    s_xc[i] = gxc[go];
  }
#endif
  __syncthreads();

  // per-image scalars (uniform)
  const float eps_n  = epsilon[n];
  const float phi_n  = phi[n];
  const float sigma  = sigma_edge[n];        // size_factor = 1024/1024 = 1
  const float half_step = 2.0f * sigma;
  const float inv2ss = 1.0f / (2.0f * sigma * sigma + FEPS);

  const int lx  = threadIdx.x & 31;
  const int ly0 = threadIdx.x >> 5;          // 0..7

  for (int r = 0; r < 4; ++r) {
    const int gx = tileX + lx;
    const int gy = tileY + ly0 + r * 8;
    const int li = (gy - w0y) * WIN + (gx - w0x);

    const float p0x = ((float)gx + 0.5f) * (1.0f / (float)W);
    const float p0y = ((float)gy + 0.5f) * (1.0f / (float)H);
    const float t0x = s_tx[li];
    const float t0y = s_ty[li];

    float Ha = s_xc[li];
    float Wa = 1.0f;

    #pragma unroll
    for (int dir = 0; dir < 2; ++dir) {
      const float sgn = (dir == 0) ? 1.0f : -1.0f;
      float px = p0x, py = p0y;
      float tpx = sgn * t0x, tpy = sgn * t0y;
      float wsum = 0.0f;
      for (int it = 0; it < MAXSTEP; ++it) {
        const float sw_old = wsum;
        // ---- lic_step ----
        Bil b = bil_idx(px, py, w0x, w0y);
        float tx = b.w00 * s_tx[b.i00] + b.w01 * s_tx[b.i01]
                 + b.w10 * s_tx[b.i10] + b.w11 * s_tx[b.i11];
        float ty = b.w00 * s_ty[b.i00] + b.w01 * s_ty[b.i01]
                 + b.w10 * s_ty[b.i10] + b.w11 * s_ty[b.i11];
        if (tx * tpx + ty * tpy < 0.0f) { tx = -tx; ty = -ty; }
        const float fracx = px - truncf(px);
        const float fracy = py - truncf(py);
        const float rx = fabsf(safediv(fracx - 0.5f - sgnf(tx), tx));
        const float ry = fabsf(safediv(fracy - 0.5f - sgnf(ty), ty));
        const float dw = (fabsf(tx) > fabsf(ty)) ? rx : ry;
        px += tx * dw * (1.0f / (float)W);
        py += ty * dw * (1.0f / (float)H);
        wsum += dw;
        // ---- accumulate ----
        float k = dw * __expf(-wsum * wsum * inv2ss);
        const bool inb = (px >= 0.0f) && (px < 1.0f) &&
                         (py >= 0.0f) && (py < 1.0f) && (sw_old < half_step);
        k = inb ? k : 0.0f;
        Bil b2 = bil_idx(px, py, w0x, w0y);
        const float xs = b2.w00 * s_xc[b2.i00] + b2.w01 * s_xc[b2.i01]
                       + b2.w10 * s_xc[b2.i10] + b2.w11 * s_xc[b2.i11];
        Ha += k * xs;
        Wa += k;
        tpx = tx; tpy = ty;
        // wsum is monotone; once >= half_step every future k is 0
        if (wsum >= half_step) break;
      }
    }

    const float Hn  = Ha / Wa;
    const float res = (Hn > eps_n) ? 0.0f
                                   : -tanhf(100.0f * phi_n * (Hn - eps_n));
    const size_t pix = (size_t)gy * W + gx;
    out[((size_t)n * 3 + 0) * HW + pix] = res;
    out[((size_t)n * 3 + 1) * HW + pix] = 0.0f;
    out[((size_t)n * 3 + 2) * HW + pix] = 0.0f;
  }
}

extern "C" void kernel_launch(void* const* d_in, const int* in_sizes, int n_in,
                              void* d_out, int out_size, void* d_ws, size_t ws_size,
                              hipStream_t stream) {
  (void)in_sizes; (void)n_in; (void)out_size; (void)d_ws; (void)ws_size;
  const float* x          = (const float*)d_in[0];
  const float* tangent    = (const float*)d_in[1];
  const float* epsilon    = (const float*)d_in[2];
  const float* sigma_edge = (const float*)d_in[3];
  const float* phi        = (const float*)d_in[4];
  float* out = (float*)d_out;
  dim3 grid(W / TILE, H / TILE, NIMG);
  dim3 block(256);
  lic_xdog_kernel<<<grid, block, 0, stream>>>(x, tangent, epsilon, sigma_edge,
                                              phi, out);
}